// DeeperGCNLayerMix_14697378087224
// MI455X (gfx1250) — compile-verified
//
#include <hip/hip_runtime.h>
#include <math.h>

#define NN 50000
#define EE 600000
#define DD 128
#define D2 256
#define EPS_MSG 1e-7f
#define BN_EPS 1e-5f
#define LN_EPS 1e-5f

typedef __attribute__((ext_vector_type(2))) float v2f;
typedef __attribute__((ext_vector_type(8))) float v8f;

// Monotonic order-preserving float->uint key (works for any sign of t).
__device__ __forceinline__ unsigned fkey(float f) {
  unsigned b = __float_as_uint(f);
  return (b & 0x80000000u) ? ~b : (b | 0x80000000u);
}
__device__ __forceinline__ float fkey_inv(unsigned k) {
  unsigned b = (k & 0x80000000u) ? (k & 0x7fffffffu) : ~k;
  return __uint_as_float(b);
}

__global__ void zero_ws_kernel(float* __restrict__ p, size_t n) {
  size_t i = (size_t)blockIdx.x * blockDim.x + threadIdx.x;
  if (i < n) p[i] = 0.0f;
}

// ---------------- Edge pass 1: per-(dst, channel) max of logits -------------
__global__ void edge_max_kernel(const float* __restrict__ x, const int* __restrict__ ei,
                                const float* __restrict__ tp, unsigned* __restrict__ mkey) {
  int wid = threadIdx.x >> 5, lane = threadIdx.x & 31;
  int e = blockIdx.x * 8 + wid;
  if (e >= EE) return;
  int src = ei[e], dst = ei[EE + e];
  float t = tp[0];
  const float4 xv = *(const float4*)(x + (size_t)src * DD + lane * 4);
  unsigned* mrow = mkey + (size_t)dst * DD + lane * 4;
  atomicMax(mrow + 0, fkey((fmaxf(xv.x, 0.f) + EPS_MSG) * t));
  atomicMax(mrow + 1, fkey((fmaxf(xv.y, 0.f) + EPS_MSG) * t));
  atomicMax(mrow + 2, fkey((fmaxf(xv.z, 0.f) + EPS_MSG) * t));
  atomicMax(mrow + 3, fkey((fmaxf(xv.w, 0.f) + EPS_MSG) * t));
}

// ---------------- Edge pass 2: sum of exp(logit - m[dst]) -------------------
__global__ void edge_expsum_kernel(const float* __restrict__ x, const int* __restrict__ ei,
                                   const float* __restrict__ tp, const unsigned* __restrict__ mkey,
                                   float* __restrict__ s) {
  int wid = threadIdx.x >> 5, lane = threadIdx.x & 31;
  int e = blockIdx.x * 8 + wid;
  if (e >= EE) return;
  int src = ei[e], dst = ei[EE + e];
  float t = tp[0];
  const float4 xv = *(const float4*)(x + (size_t)src * DD + lane * 4);
  size_t doff = (size_t)dst * DD + lane * 4;
  const unsigned* mrow = mkey + doff;
  float* srow = s + doff;
  float v[4] = {xv.x, xv.y, xv.z, xv.w};
#pragma unroll
  for (int j = 0; j < 4; ++j) {
    float lg = (fmaxf(v[j], 0.f) + EPS_MSG) * t;
    unsigned k = mrow[j];
    float m = (k == 0u) ? 0.f : fkey_inv(k);
    atomicAdd(srow + j, expf(lg - m));
  }
}

// ---------------- Edge pass 3: agg[dst] += msg * alpha ----------------------
__global__ void edge_agg_kernel(const float* __restrict__ x, const int* __restrict__ ei,
                                const float* __restrict__ tp, const unsigned* __restrict__ mkey,
                                const float* __restrict__ s, float* __restrict__ agg) {
  int wid = threadIdx.x >> 5, lane = threadIdx.x & 31;
  int e = blockIdx.x * 8 + wid;
  if (e >= EE) return;
  int src = ei[e], dst = ei[EE + e];
  float t = tp[0];
  const float4 xv = *(const float4*)(x + (size_t)src * DD + lane * 4);
  size_t doff = (size_t)dst * DD + lane * 4;
  const unsigned* mrow = mkey + doff;
  const float* srow = s + doff;
  float* arow = agg + doff;
  float v[4] = {xv.x, xv.y, xv.z, xv.w};
#pragma unroll
  for (int j = 0; j < 4; ++j) {
    float msg = fmaxf(v[j], 0.f) + EPS_MSG;
    unsigned k = mrow[j];
    float m = (k == 0u) ? 0.f : fkey_inv(k);
    float ex = expf(msg * t - m);
    float alpha = ex / (srow[j] + 1e-16f);
    atomicAdd(arow + j, msg * alpha);
  }
}

// ---------------- GEMM1: h = (agg + x) @ W1 + b1  [N x 256] -----------------
// One wave computes a 16x64 strip with V_WMMA_F32_16X16X4_F32.
// A frag (16x4 f32): lane ml (0..15) = row M; lane half kh selects K pair.
// B/C/D: lane = column; vgpr index (+8*kh for C/D) = row.
// Rows are CLAMPED (not predicated): OOB lanes compute rows never stored.
__global__ void gemm1_kernel(const float* __restrict__ x, const float* __restrict__ agg,
                             const float* __restrict__ W1, const float* __restrict__ b1,
                             float* __restrict__ h) {
  int wid = threadIdx.x >> 5, lane = threadIdx.x & 31;
  int ml = lane & 15, kh = lane >> 4;
  int m0 = (blockIdx.x * 8 + wid) * 16;
  int n0 = blockIdx.y * 64;
  if (m0 >= NN) return;
  int row = m0 + ml;
  row = row < NN ? row : NN - 1;  // clamp: keeps loads unconditional
  size_t arow = (size_t)row * DD;
  v8f c0 = {}, c1 = {}, c2 = {}, c3 = {};
  for (int k = 0; k < DD; k += 4) {
    int ka = k + 2 * kh;
    const float2 ag = *(const float2*)(agg + arow + ka);
    const float2 xr = *(const float2*)(x + arow + ka);
    v2f a;
    a.x = ag.x + xr.x;
    a.y = ag.y + xr.y;
    const float* w0 = W1 + (size_t)ka * D2 + n0 + ml;
    const float* w1 = W1 + (size_t)(ka + 1) * D2 + n0 + ml;
    v2f b;
    b.x = w0[0];  b.y = w1[0];
    c0 = __builtin_amdgcn_wmma_f32_16x16x4_f32(false, a, false, b, (short)0, c0, false, false);
    b.x = w0[16]; b.y = w1[16];
    c1 = __builtin_amdgcn_wmma_f32_16x16x4_f32(false, a, false, b, (short)0, c1, false, false);
    b.x = w0[32]; b.y = w1[32];
    c2 = __builtin_amdgcn_wmma_f32_16x16x4_f32(false, a, false, b, (short)0, c2, false, false);
    b.x = w0[48]; b.y = w1[48];
    c3 = __builtin_amdgcn_wmma_f32_16x16x4_f32(false, a, false, b, (short)0, c3, false, false);
  }
  float bb0 = b1[n0 + ml + 0], bb1 = b1[n0 + ml + 16];
  float bb2 = b1[n0 + ml + 32], bb3 = b1[n0 + ml + 48];
  if (m0 + 16 <= NN) {
#pragma unroll
    for (int r = 0; r < 8; ++r) {
      size_t base = (size_t)(m0 + r + 8 * kh) * D2 + n0 + ml;
      h[base + 0]  = c0[r] + bb0;
      h[base + 16] = c1[r] + bb1;
      h[base + 32] = c2[r] + bb2;
      h[base + 48] = c3[r] + bb3;
    }
  } else {
#pragma unroll
    for (int r = 0; r < 8; ++r) {
      int orow = m0 + r + 8 * kh;
      if (orow < NN) {
        size_t base = (size_t)orow * D2 + n0 + ml;
        h[base + 0]  = c0[r] + bb0;
        h[base + 16] = c1[r] + bb1;
        h[base + 32] = c2[r] + bb2;
        h[base + 48] = c3[r] + bb3;
      }
    }
  }
}

// ---------------- BatchNorm stats over N rows (coalesced) -------------------
__global__ void bnstat_kernel(const float* __restrict__ h, float* __restrict__ bnsum,
                              float* __restrict__ bnsq) {
  int col = threadIdx.x;  // 256 threads = 256 channels
  float s = 0.f, q = 0.f;
  for (int r = blockIdx.x; r < NN; r += gridDim.x) {
    float v = h[(size_t)r * D2 + col];
    s += v; q += v * v;
  }
  atomicAdd(&bnsum[col], s);
  atomicAdd(&bnsq[col], q);
}

__global__ void bnfinalize_kernel(const float* __restrict__ bnsum, const float* __restrict__ bnsq,
                                  const float* __restrict__ g, const float* __restrict__ b,
                                  float* __restrict__ scale, float* __restrict__ shift) {
  int col = threadIdx.x;
  float inv_n = 1.0f / (float)NN;
  float mu = bnsum[col] * inv_n;
  float var = fmaxf(bnsq[col] * inv_n - mu * mu, 0.f);
  float sc = g[col] * rsqrtf(var + BN_EPS);
  scale[col] = sc;
  shift[col] = b[col] - mu * sc;
}

// ---------------- GEMM2: y = relu(BN(h)) @ W2 + b2  [N x 128] ---------------
__global__ void gemm2_kernel(const float* __restrict__ h, const float* __restrict__ scale,
                             const float* __restrict__ shift, const float* __restrict__ W2,
                             const float* __restrict__ b2, float* __restrict__ y) {
  int wid = threadIdx.x >> 5, lane = threadIdx.x & 31;
  int ml = lane & 15, kh = lane >> 4;
  int m0 = (blockIdx.x * 8 + wid) * 16;
  int n0 = blockIdx.y * 64;
  if (m0 >= NN) return;
  int row = m0 + ml;
  row = row < NN ? row : NN - 1;  // clamp
  size_t hrow = (size_t)row * D2;
  v8f c0 = {}, c1 = {}, c2 = {}, c3 = {};
  for (int k = 0; k < D2; k += 4) {
    int ka = k + 2 * kh;
    const float2 hv = *(const float2*)(h + hrow + ka);
    const float2 sc = *(const float2*)(scale + ka);
    const float2 sh = *(const float2*)(shift + ka);
    v2f a;
    a.x = fmaxf(hv.x * sc.x + sh.x, 0.f);
    a.y = fmaxf(hv.y * sc.y + sh.y, 0.f);
    const float* w0 = W2 + (size_t)ka * DD + n0 + ml;
    const float* w1 = W2 + (size_t)(ka + 1) * DD + n0 + ml;
    v2f b;
    b.x = w0[0];  b.y = w1[0];
    c0 = __builtin_amdgcn_wmma_f32_16x16x4_f32(false, a, false, b, (short)0, c0, false, false);
    b.x = w0[16]; b.y = w1[16];
    c1 = __builtin_amdgcn_wmma_f32_16x16x4_f32(false, a, false, b, (short)0, c1, false, false);
    b.x = w0[32]; b.y = w1[32];
    c2 = __builtin_amdgcn_wmma_f32_16x16x4_f32(false, a, false, b, (short)0, c2, false, false);
    b.x = w0[48]; b.y = w1[48];
    c3 = __builtin_amdgcn_wmma_f32_16x16x4_f32(false, a, false, b, (short)0, c3, false, false);
  }
  float bb0 = b2[n0 + ml + 0], bb1 = b2[n0 + ml + 16];
  float bb2 = b2[n0 + ml + 32], bb3 = b2[n0 + ml + 48];
  if (m0 + 16 <= NN) {
#pragma unroll
    for (int r = 0; r < 8; ++r) {
      size_t base = (size_t)(m0 + r + 8 * kh) * DD + n0 + ml;
      y[base + 0]  = c0[r] + bb0;
      y[base + 16] = c1[r] + bb1;
      y[base + 32] = c2[r] + bb2;
      y[base + 48] = c3[r] + bb3;
    }
  } else {
#pragma unroll
    for (int r = 0; r < 8; ++r) {
      int orow = m0 + r + 8 * kh;
      if (orow < NN) {
        size_t base = (size_t)orow * DD + n0 + ml;
        y[base + 0]  = c0[r] + bb0;
        y[base + 16] = c1[r] + bb1;
        y[base + 32] = c2[r] + bb2;
        y[base + 48] = c3[r] + bb3;
      }
    }
  }
}

// ---------------- LayerNorm + mixed activation + residual -------------------
__global__ void ln_kernel(const float* __restrict__ y, const float* __restrict__ x,
                          const float* __restrict__ g, const float* __restrict__ b,
                          float* __restrict__ out) {
  int wid = threadIdx.x >> 5, lane = threadIdx.x & 31;
  int row = blockIdx.x * 8 + wid;
  if (row >= NN) return;
  size_t base = (size_t)row * DD + lane * 4;
  const float4 yv = *(const float4*)(y + base);
  float s = yv.x + yv.y + yv.z + yv.w;
#pragma unroll
  for (int off = 16; off > 0; off >>= 1) s += __shfl_xor(s, off, 32);
  float mu = s * (1.0f / DD);
  float dx = yv.x - mu, dy = yv.y - mu, dz = yv.z - mu, dw = yv.w - mu;
  float q = dx * dx + dy * dy + dz * dz + dw * dw;
#pragma unroll
  for (int off = 16; off > 0; off >>= 1) q += __shfl_xor(q, off, 32);
  float rstd = rsqrtf(q * (1.0f / DD) + LN_EPS);
  const float4 xv = *(const float4*)(x + base);
  int c = lane * 4;
  float z0 = dx * rstd * g[c + 0] + b[c + 0];
  float z1 = dy * rstd * g[c + 1] + b[c + 1];
  float z2 = dz * rstd * g[c + 2] + b[c + 2];
  float z3 = dw * rstd * g[c + 3] + b[c + 3];
  float4 o;
  o.x = 0.5f * xv.x + 0.5f * fmaxf(z0, 0.f) + 0.5f * z0;
  o.y = 0.5f * xv.y + 0.5f * fmaxf(z1, 0.f) + 0.5f * z1;
  o.z = 0.5f * xv.z + 0.5f * fmaxf(z2, 0.f) + 0.5f * z2;
  o.w = 0.5f * xv.w + 0.5f * fmaxf(z3, 0.f) + 0.5f * z3;
  *(float4*)(out + base) = o;
}

extern "C" void kernel_launch(void* const* d_in, const int* in_sizes, int n_in,
                              void* d_out, int out_size, void* d_ws, size_t ws_size,
                              hipStream_t stream) {
  (void)in_sizes; (void)n_in; (void)out_size; (void)ws_size;
  const float* x   = (const float*)d_in[0];
  const int*   ei  = (const int*)d_in[1];
  const float* t   = (const float*)d_in[2];
  const float* W1  = (const float*)d_in[3];
  const float* b1  = (const float*)d_in[4];
  const float* bng = (const float*)d_in[5];
  const float* bnb = (const float*)d_in[6];
  const float* W2  = (const float*)d_in[7];
  const float* b2  = (const float*)d_in[8];
  const float* lng = (const float*)d_in[9];
  const float* lnb = (const float*)d_in[10];
  float* out = (float*)d_out;
  float* ws  = (float*)d_ws;

  size_t ND = (size_t)NN * DD;
  unsigned* mkey  = (unsigned*)ws;          // N*D keys (zero = empty)
  float*    sbuf  = ws + ND;                // N*D exp-sums
  float*    agg   = ws + 2 * ND;            // N*D aggregation
  float*    bnsum = ws + 3 * ND;            // 256
  float*    bnsq  = bnsum + D2;             // 256
  float*    h     = bnsq + D2;              // N*256 pre-BN hidden
  float*    scale = h + (size_t)NN * D2;    // 256
  float*    shift = scale + D2;             // 256
  float*    y     = shift + D2;             // N*128 pre-LN output

  size_t nzero = 3 * ND + 2 * D2;  // mkey + sbuf + agg + bn partials
  zero_ws_kernel<<<(unsigned)((nzero + 255) / 256), 256, 0, stream>>>(ws, nzero);

  int eblocks = (EE + 7) / 8;
  edge_max_kernel<<<eblocks, 256, 0, stream>>>(x, ei, t, mkey);
  edge_expsum_kernel<<<eblocks, 256, 0, stream>>>(x, ei, t, mkey, sbuf);
  edge_agg_kernel<<<eblocks, 256, 0, stream>>>(x, ei, t, mkey, sbuf, agg);

  dim3 g1((NN + 127) / 128, 4);
  gemm1_kernel<<<g1, 256, 0, stream>>>(x, agg, W1, b1, h);

  bnstat_kernel<<<512, 256, 0, stream>>>(h, bnsum, bnsq);
  bnfinalize_kernel<<<1, 256, 0, stream>>>(bnsum, bnsq, bng, bnb, scale, shift);

  dim3 g2((NN + 127) / 128, 2);
  gemm2_kernel<<<g2, 256, 0, stream>>>(h, scale, shift, W2, b2, y);

  ln_kernel<<<(NN + 7) / 8, 256, 0, stream>>>(y, x, lng, lnb, out);
}